// GCN_76201309766159
// MI455X (gfx1250) — compile-verified
//
#include <hip/hip_runtime.h>
#include <hip/hip_bf16.h>

// ---------------------------------------------------------------------------
// GCN layer for MI455X (gfx1250, wave32).
//   feat = norm_src * (x @ W)          -> fp32 WMMA 16x16x4, row-scale at epilogue
//   agg  = segment_sum(feat[src], dst) -> L2-resident f32 atomics
//   h    = relu(agg * norm_dst + b); h2 = (h >= 0.5)
// ---------------------------------------------------------------------------

typedef __attribute__((ext_vector_type(2))) float v2f;
typedef __attribute__((ext_vector_type(8))) float v8f;

static constexpr int N_NODES  = 50000;
static constexpr int N_EDGES  = 1600000;
static constexpr int IN_FEATS = 256;
static constexpr int H_FEATS  = 64;

// ----------------------------- zero fill -----------------------------------
__global__ void gcn_zero_f32(float* __restrict__ p, int n) {
    int i = blockIdx.x * blockDim.x + threadIdx.x;
    if (i < n) p[i] = 0.0f;
}

// ----------------------------- degrees -------------------------------------
__global__ void gcn_degrees(const int* __restrict__ src, const int* __restrict__ dst,
                            unsigned* __restrict__ out_deg, unsigned* __restrict__ in_deg,
                            int n_edges) {
    int e = blockIdx.x * blockDim.x + threadIdx.x;
    if (e < n_edges) {
        atomicAdd(&out_deg[src[e]], 1u);
        atomicAdd(&in_deg[dst[e]], 1u);
    }
}

// ----------------------------- norms ---------------------------------------
__global__ void gcn_norms(const unsigned* __restrict__ out_deg,
                          const unsigned* __restrict__ in_deg,
                          float* __restrict__ norm_src, float* __restrict__ norm_dst,
                          int n_nodes) {
    int i = blockIdx.x * blockDim.x + threadIdx.x;
    if (i < n_nodes) {
        float od = fmaxf((float)out_deg[i], 1.0f);
        float id = fmaxf((float)in_deg[i], 1.0f);
        norm_src[i] = 1.0f / sqrtf(od);
        norm_dst[i] = 1.0f / sqrtf(id);
    }
}

// --------------- fp32 WMMA GEMM: feat = diag(norm_src) * (x @ W) ------------
// Grid: 3125 blocks (one 16-row M tile each), block = 128 threads = 4 waves.
// Wave w computes the 16x16 tile at columns [16w, 16w+16).  K = 256 = 64 x 4.
// Row scaling commutes with the GEMM, so it is applied to the D rows at the
// epilogue -- keeps the k-loop a pure load->wmma stream (no VALU on the
// critical path).
//
// VGPR layouts (CDNA5 ISA 7.12.2, 32-bit data):
//   A 16x4 : lanes 0-15 hold M=lane, {K, K+1}; lanes 16-31 hold M=lane-16, {K+2, K+3}
//   B 4x16 : lanes 0-15 hold N=lane, {K, K+1}; lanes 16-31 hold N=lane-16, {K+2, K+3}
//   C/D    : VGPR r -> M = r + 8*(lane>=16), N = lane & 15
__global__ void gcn_gemm_wmma(const float* __restrict__ x, const float* __restrict__ W,
                              const float* __restrict__ norm_src,
                              float* __restrict__ feat) {
    const int wave  = threadIdx.x >> 5;   // 0..3 -> N tile
    const int lane  = threadIdx.x & 31;
    const int l15   = lane & 15;
    const int lhalf = lane >> 4;          // 0 or 1
    const int mtile = blockIdx.x;         // 0..3124

    const int row = mtile * 16 + l15;     // A row held by this lane
    const int col = wave * 16 + l15;      // B/D column held by this lane

    const float* __restrict__ xrow = x + (size_t)row * IN_FEATS;

    v8f acc = {0.f, 0.f, 0.f, 0.f, 0.f, 0.f, 0.f, 0.f};

#pragma unroll 8
    for (int k = 0; k < IN_FEATS; k += 4) {
        const int ka = k + 2 * lhalf;
        v2f a, b;
        a.x = xrow[ka];
        a.y = xrow[ka + 1];
        b.x = W[(size_t)ka * H_FEATS + col];
        b.y = W[(size_t)(ka + 1) * H_FEATS + col];
        // D = A x B + C   (fp32, 16x16x4)
        acc = __builtin_amdgcn_wmma_f32_16x16x4_f32(
            /*neg_a=*/false, a, /*neg_b=*/false, b,
            /*c_mod=*/(short)0, acc, /*reuse_a=*/false, /*reuse_b=*/false);
    }

    // Epilogue: scale D rows by norm_src and store.
    const int baserow = mtile * 16 + 8 * lhalf;
#pragma unroll
    for (int r = 0; r < 8; ++r) {
        const float ns = norm_src[baserow + r];
        feat[(size_t)(baserow + r) * H_FEATS + col] = acc[r] * ns;
    }
}

// --------------------- edge scatter: agg[dst] += feat[src] ------------------
// 16 lanes per edge, 4 floats per lane (float4 gather; feat & agg are L2-hot).
__global__ void gcn_scatter(const float* __restrict__ feat,
                            const int* __restrict__ src, const int* __restrict__ dst,
                            float* __restrict__ agg, int n_edges) {
    const int gid = blockIdx.x * blockDim.x + threadIdx.x;
    const int e = gid >> 4;
    if (e >= n_edges) return;
    const int quad = (gid & 15) * 4;

    const int s = src[e];
    const int d = dst[e];
    const float4 v = *(const float4*)(feat + (size_t)s * H_FEATS + quad);
    float* out = agg + (size_t)d * H_FEATS + quad;
    atomicAdd(out + 0, v.x);
    atomicAdd(out + 1, v.y);
    atomicAdd(out + 2, v.z);
    atomicAdd(out + 3, v.w);
}

// --------------------- finalize: relu + threshold (in place) ----------------
__global__ void gcn_finalize(float* __restrict__ out,            // [0..N*H): agg -> h
                             const float* __restrict__ norm_dst,
                             const float* __restrict__ bias,
                             int n_nodes) {
    const int gid = blockIdx.x * blockDim.x + threadIdx.x;
    const int total = n_nodes * H_FEATS;
    if (gid >= total) return;
    const int node = gid >> 6;      // /64
    const int hch  = gid & 63;
    float v = out[gid] * norm_dst[node] + bias[hch];
    v = fmaxf(v, 0.0f);
    out[gid] = v;                                   // h
    out[gid + total] = (v >= 0.5f) ? 1.0f : 0.0f;   // h_clone
}

// ---------------------------------------------------------------------------
extern "C" void kernel_launch(void* const* d_in, const int* in_sizes, int n_in,
                              void* d_out, int out_size, void* d_ws, size_t ws_size,
                              hipStream_t stream) {
    const float* x   = (const float*)d_in[0];
    const int*   src = (const int*)  d_in[1];
    const int*   dst = (const int*)  d_in[2];
    const float* W   = (const float*)d_in[3];
    const float* b   = (const float*)d_in[4];

    float* out = (float*)d_out;   // [0, N*H): h (also the agg accumulator); [N*H, 2*N*H): h_clone

    // Workspace layout (16B-aligned offsets):
    //   [0,      200000)  out_deg  u32[50000]
    //   [200000, 400000)  in_deg   u32[50000]
    //   [400000, 600000)  norm_src f32[50000]
    //   [600000, 800000)  norm_dst f32[50000]
    //   [800000, 800000+12.8MB)  feat f32[50000*64]
    char* ws = (char*)d_ws;
    unsigned* out_deg  = (unsigned*)(ws + 0);
    unsigned* in_deg   = (unsigned*)(ws + 200000);
    float*    norm_src = (float*)   (ws + 400000);
    float*    norm_dst = (float*)   (ws + 600000);
    float*    feat     = (float*)   (ws + 800000);

    const int NH = N_NODES * H_FEATS;   // 3,200,000

    // 1) zero degree counters (u32 zero == f32 zero bit pattern) and agg region
    gcn_zero_f32<<<(2 * N_NODES + 255) / 256, 256, 0, stream>>>((float*)out_deg, 2 * N_NODES);
    gcn_zero_f32<<<(NH + 255) / 256, 256, 0, stream>>>(out, NH);

    // 2) degrees
    gcn_degrees<<<(N_EDGES + 255) / 256, 256, 0, stream>>>(src, dst, out_deg, in_deg, N_EDGES);

    // 3) norms
    gcn_norms<<<(N_NODES + 255) / 256, 256, 0, stream>>>(out_deg, in_deg, norm_src, norm_dst, N_NODES);

    // 4) fp32 WMMA GEMM: feat = diag(norm_src) * (x @ W)
    gcn_gemm_wmma<<<N_NODES / 16, 128, 0, stream>>>(x, W, norm_src, feat);

    // 5) edge scatter-add into out[0..NH)
    {
        const long long threads = (long long)N_EDGES * 16;
        const int blocks = (int)((threads + 255) / 256);
        gcn_scatter<<<blocks, 256, 0, stream>>>(feat, src, dst, out, N_EDGES);
    }

    // 6) finalize: h = relu(agg*norm_dst + b), h_clone = (h >= 0.5)
    gcn_finalize<<<(NH + 255) / 256, 256, 0, stream>>>(out, norm_dst, b, N_NODES);
}